// GlobalPointer_65292092833908
// MI455X (gfx1250) — compile-verified
//
#include <hip/hip_runtime.h>
#include <math.h>

#define HEADS 12
#define HEAD_SIZE 64
#define HIDDEN 768
#define EDIM (2 * HEADS * HEAD_SIZE)   // 1536
#define BATCH 8
#define SEQ 1024
#define NEGV 1000000000000.0f

typedef __attribute__((ext_vector_type(2))) float v2f;
typedef __attribute__((ext_vector_type(8))) float v8f;

// ---------------------------------------------------------------------------
// Kernel 1: proj = x @ W + b, then interleaved RoPE on q/k halves.
// Each wave owns a 64x64 output region: 4x4 grid of 16x16 f32 accumulators.
// K = 768 -> 192 k-steps of (4 A-frag loads + 4 B-frag loads + 16 WMMAs),
// giving 4x less L2 traffic than 16x16-per-wave blocking.
// A 64-aligned column block never crosses a head (128) or q/k (64) boundary,
// so head index and q/k side are wave-uniform in the epilogue.
// ---------------------------------------------------------------------------
__global__ __launch_bounds__(128) void proj_rope_kernel(
    const float* __restrict__ x,      // (8192, 768)
    const float* __restrict__ W,      // (768, 1536)
    const float* __restrict__ bias,   // (1536,)
    float* __restrict__ q_ws,         // (8,12,1024,64)
    float* __restrict__ k_ws)         // (8,12,1024,64)
{
    const int lane  = threadIdx.x & 31;
    const int wave  = threadIdx.x >> 5;
    const int tile  = blockIdx.x * 4 + wave;     // 3072 wave-tiles total
    const int tiles_n = EDIM / 64;               // 24
    const int m0 = (tile / tiles_n) * 64;        // row block in [0,8192)
    const int n0 = (tile % tiles_n) * 64;        // col block in [0,1536)

    const int lhalf = lane >> 4;                 // 0/1 -> K sub-pair select
    const int l15   = lane & 15;

    v8f acc[4][4] = {};
    for (int k = 0; k < HIDDEN; k += 4) {
        const int kb = k + lhalf * 2;
        v2f af[4], bf[4];
        #pragma unroll
        for (int i = 0; i < 4; ++i) {
            // A: x rows (M = m0 + i*16 + l15), 2 consecutive K vals (8B aligned)
            af[i] = *(const v2f*)(x + (size_t)(m0 + i * 16 + l15) * HIDDEN + kb);
            // B: W columns (N = n0 + i*16 + l15), rows kb, kb+1
            const int n = n0 + i * 16 + l15;
            bf[i].x = W[(size_t)kb * EDIM + n];
            bf[i].y = W[(size_t)(kb + 1) * EDIM + n];
        }
        #pragma unroll
        for (int i = 0; i < 4; ++i)
            #pragma unroll
            for (int j = 0; j < 4; ++j)
                acc[i][j] = __builtin_amdgcn_wmma_f32_16x16x4_f32(
                    false, af[i], false, bf[j], (short)0, acc[i][j], false, false);
    }

    // -------- epilogue: bias + RoPE + scatter --------
    const int  h    = n0 >> 7;                   // head (wave-uniform)
    const bool is_k = (n0 & 64) != 0;            // q/k side (wave-uniform)
    float* __restrict__ dst = is_k ? k_ws : q_ws;

    #pragma unroll
    for (int j = 0; j < 4; ++j) {
        const int Ncol = n0 + j * 16 + l15;      // global column
        const int d    = (j * 16 + l15);         // head-dim index (0..63)
        const float bv = bias[Ncol];
        // interleaved rotate_half: even d -> -partner, odd d -> +partner
        const float sgn = (d & 1) ? 1.0f : -1.0f;
        // inv_freq = 10000^(-2*(d/2)/64)
        const float inv = __expf(-(float)(d & ~1) * (logf(10000.0f) / 64.0f));
        #pragma unroll
        for (int i = 0; i < 4; ++i) {
            #pragma unroll
            for (int r = 0; r < 8; ++r) {
                const int Mr  = m0 + i * 16 + r + lhalf * 8;  // row in [0,8192)
                const int bb  = Mr >> 10;                     // batch
                const int pos = Mr & (SEQ - 1);               // sequence pos
                const float v = acc[i][j][r] + bv;            // biased proj
                const float p = __shfl_xor(v, 1, 32);         // RoPE partner
                float s, ct;
                __sincosf((float)pos * inv, &s, &ct);
                const float outv = v * ct + sgn * p * s;
                dst[(((size_t)(bb * HEADS + h) * SEQ + pos) * HEAD_SIZE) + d] = outv;
            }
        }
    }
}

// ---------------------------------------------------------------------------
// Kernel 2: logits[b,h] = q @ k^T with mask + causal tril + 1/sqrt(64) scale.
// Each wave owns a 64x64 output region: 4x4 grid of 16x16 f32 accumulators,
// K = 64 -> 16 k-steps of (4 A loads + 4 B loads + 16 WMMAs).
// ---------------------------------------------------------------------------
__global__ __launch_bounds__(128) void attn_logits_kernel(
    const float* __restrict__ q_ws,   // (8,12,1024,64)
    const float* __restrict__ k_ws,   // (8,12,1024,64)
    const float* __restrict__ mask,   // (8,1024)
    float* __restrict__ out)          // (8,12,1024,1024)
{
    const int lane = threadIdx.x & 31;
    const int wave = threadIdx.x >> 5;
    const int t    = blockIdx.x * 4 + wave;   // 24576 wave-tiles
    const int bh   = t >> 8;                  // (b*12 + h)
    const int tile = t & 255;
    const int b    = bh / HEADS;
    const int tm   = (tile >> 4) * 64;        // query-row block
    const int tn   = (tile & 15) * 64;        // key-col block

    const float* __restrict__ qbase = q_ws + (size_t)bh * SEQ * HEAD_SIZE;
    const float* __restrict__ kbase = k_ws + (size_t)bh * SEQ * HEAD_SIZE;

    const int lhalf = lane >> 4;
    const int l15   = lane & 15;

    v8f acc[4][4] = {};
    for (int k = 0; k < HEAD_SIZE; k += 4) {
        const int kb = k + lhalf * 2;
        v2f af[4], bf[4];
        #pragma unroll
        for (int i = 0; i < 4; ++i) {
            // A: q rows (M = tm + i*16 + l15); B: k rows act as columns (N)
            af[i] = *(const v2f*)(qbase + (size_t)(tm + i * 16 + l15) * HEAD_SIZE + kb);
            bf[i] = *(const v2f*)(kbase + (size_t)(tn + i * 16 + l15) * HEAD_SIZE + kb);
        }
        #pragma unroll
        for (int i = 0; i < 4; ++i)
            #pragma unroll
            for (int j = 0; j < 4; ++j)
                acc[i][j] = __builtin_amdgcn_wmma_f32_16x16x4_f32(
                    false, af[i], false, bf[j], (short)0, acc[i][j], false, false);
    }

    // -------- epilogue: masks, causal tril(k=-1), scale, store --------
    float mkv[4];
    #pragma unroll
    for (int j = 0; j < 4; ++j)
        mkv[j] = mask[b * SEQ + tn + j * 16 + l15];

    float* __restrict__ obase = out + (size_t)bh * SEQ * SEQ;
    #pragma unroll
    for (int i = 0; i < 4; ++i) {
        #pragma unroll
        for (int r = 0; r < 8; ++r) {
            const int m  = tm + i * 16 + r + lhalf * 8;
            const float mq = mask[b * SEQ + m];
            #pragma unroll
            for (int j = 0; j < 4; ++j) {
                const int n = tn + j * 16 + l15;
                float v = acc[i][j][r];
                v = v * mq     + NEGV * (mq     - 1.0f);
                v = v * mkv[j] + NEGV * (mkv[j] - 1.0f);
                if (n < m) v -= NEGV;           // tril(k=-1) * NEG
                v *= 0.125f;                    // 1/sqrt(64)
                obase[(size_t)m * SEQ + n] = v;
            }
        }
    }
}

// ---------------------------------------------------------------------------
extern "C" void kernel_launch(void* const* d_in, const int* in_sizes, int n_in,
                              void* d_out, int out_size, void* d_ws, size_t ws_size,
                              hipStream_t stream) {
    const float* x    = (const float*)d_in[0];   // (8,1024,768)
    const float* W    = (const float*)d_in[1];   // (768,1536)
    const float* bias = (const float*)d_in[2];   // (1536,)
    const float* mask = (const float*)d_in[3];   // (8,1024)
    float*       out  = (float*)d_out;           // (8,12,1024,1024)

    // Workspace: q_ws and k_ws, each 8*12*1024*64 floats = 25.2 MB
    const size_t qk_elems = (size_t)BATCH * HEADS * SEQ * HEAD_SIZE;
    float* q_ws = (float*)d_ws;
    float* k_ws = q_ws + qk_elems;

    // Kernel 1: (8192/64) * (1536/64) = 3072 wave-tiles, 4 waves per block
    const int tiles1 = (BATCH * SEQ / 64) * (EDIM / 64);
    proj_rope_kernel<<<tiles1 / 4, 128, 0, stream>>>(x, W, bias, q_ws, k_ws);

    // Kernel 2: 96 (b,h) * 16*16 region-tiles = 24576 wave-tiles
    const int tiles2 = BATCH * HEADS * (SEQ / 64) * (SEQ / 64);
    attn_logits_kernel<<<tiles2 / 4, 128, 0, stream>>>(q_ws, k_ws, mask, out);
}